// UVUTensorProduct_14920716386688
// MI455X (gfx1250) — compile-verified
//
#include <hip/hip_runtime.h>
#include <hip/hip_bf16.h>

typedef float v2f __attribute__((ext_vector_type(2)));
typedef float v8f __attribute__((ext_vector_type(8)));

#define TILE_M 16
#define LDSTR  66      // LDS row stride in floats: 8B-aligned, bank-skewed
#define WPB    4       // waves (16-row tiles) per block

__device__ __forceinline__ v8f wmma4(v2f a, v2f b, v8f c) {
  // D = A(16x4,f32) * B(4x16,f32) + C(16x16,f32)
  return __builtin_amdgcn_wmma_f32_16x16x4_f32(false, a, false, b, (short)0, c, false, false);
}

// B-fragment of 64x64 row-major W from GLOBAL: K-chunk k (rows 4k..4k+3), N-tile j.
// lane = N within tile; lanes 0-15 carry K=4k,4k+1; lanes 16-31 carry K=4k+2,4k+3
__device__ __forceinline__ v2f bfrag_g(const float* __restrict__ W, int k, int j, int lane) {
  int nn = lane & 15;
  int u  = 4 * k + ((lane >> 4) << 1);
  const float* p = W + u * 64 + j * 16 + nn;
  v2f b;
  b.x = p[0];
  b.y = p[64];
  return b;
}

// B-fragment from LDS (fragment-ordered weight copy): single ds_load_b64,
// lanes read consecutive 8B -> bank-conflict-free.
__device__ __forceinline__ v2f bfrag_l(const float* base, int k, int j, int lane) {
  const v2f* p = (const v2f*)base;
  return p[(k * 4 + j) * 32 + lane];
}

__global__ __launch_bounds__(WPB * 32) void fused_equivariant_tp(
    const float* __restrict__ x1,  const float* __restrict__ edge,
    const float* __restrict__ tpw, const float* __restrict__ nattr,
    const float* __restrict__ W0s, const float* __restrict__ W0v,
    const float* __restrict__ Wscs, const float* __restrict__ Wscv,
    const float* __restrict__ WA,  const float* __restrict__ WD,
    const float* __restrict__ WB,  const float* __restrict__ WC,
    float* __restrict__ out, int nrows, int ntiles)
{
  // Fragment-ordered copies of the 4 hot weights (used by 768/1024 WMMAs per tile)
  __shared__ __align__(16) float wlds[4][64 * 32 * 2];          // 64 KB
  // Per-wave intermediates: s0, v0_x, v0_y, v0_z (row-major, padded)
  __shared__ __align__(16) float ldsI[WPB][4][TILE_M * LDSTR];  // 67.6 KB

  const int wave = threadIdx.x >> 5;
  const int lane = threadIdx.x & 31;

  // ---- cooperative fill of frag-ordered weights: W0v, Wscv, WB, WC ----
  {
    for (int w = 0; w < 4; ++w) {
      const float* src = (w == 0) ? W0v : (w == 1) ? Wscv : (w == 2) ? WB : WC;
      float2* dst = (float2*)&wlds[w][0];
      for (int idx = threadIdx.x; idx < 2048; idx += WPB * 32) {
        int kj = idx >> 5, ln = idx & 31;
        int u = ((kj >> 2) << 2) + ((ln >> 4) << 1);   // 4k + 2*hi
        int c = ((kj & 3) << 4) + (ln & 15);           // 16j + n
        float2 t;
        t.x = src[u * 64 + c];
        t.y = src[u * 64 + 64 + c];
        dst[idx] = t;
      }
    }
  }

  int tile = blockIdx.x * WPB + wave;
  if (tile >= ntiles) tile = ntiles - 1;          // keep waves alive for barriers
  const int r0 = tile * TILE_M;
  const int n  = lane & 15;                        // A-layout row / B-layout N
  const int hi = lane >> 4;
  int row = r0 + n; if (row >= nrows) row = nrows - 1;

  const float* xrow = x1  + (size_t)row * 256;
  const float* trow = tpw + (size_t)row * 256;
  const float a   = nattr[row];
  const float e0  = edge[row * 4 + 0];
  const float e1x = edge[row * 4 + 1];
  const float e1y = edge[row * 4 + 2];
  const float e1z = edge[row * 4 + 3];

  const float c1 = 0.125f;                  // 1/sqrt(64)
  const float c2 = 0.088388347648318447f;   // 1/sqrt(128)
  const float k3 = 0.57735026918962576f;    // 1/sqrt(3)
  const float ca = c1 * a;

  float* bs0 = &ldsI[wave][0][0];
  float* bv0 = &ldsI[wave][1][0];
  float* bv1 = &ldsI[wave][2][0];
  float* bv2 = &ldsI[wave][3][0];
  const float* fW0v  = &wlds[0][0];
  const float* fWscv = &wlds[1][0];
  const float* fWB   = &wlds[2][0];
  const float* fWC   = &wlds[3][0];

  __syncthreads();   // weight staging visible to all waves

  // ---------------- stage 1: s0 = (c1*a*s) @ W_lin0_s -> LDS ----------------
  {
    v2f sf[16];
    #pragma unroll
    for (int k = 0; k < 16; ++k) {
      int u = 4 * k + 2 * hi;
      sf[k].x = ca * xrow[u];
      sf[k].y = ca * xrow[u + 1];
    }
    for (int j = 0; j < 4; ++j) {
      v8f acc = {};
      #pragma unroll
      for (int k = 0; k < 16; ++k) acc = wmma4(sf[k], bfrag_g(W0s, k, j, lane), acc);
      #pragma unroll
      for (int r = 0; r < 8; ++r) bs0[(r + 8 * hi) * LDSTR + j * 16 + n] = acc[r];
    }
  }
  // ---------------- stage 1: v0_m = (c1*a*v_m) @ W_lin0_v -> LDS ------------
  for (int m = 0; m < 3; ++m) {
    v2f vf[16];
    #pragma unroll
    for (int k = 0; k < 16; ++k) {
      int u = 4 * k + 2 * hi;
      vf[k].x = ca * xrow[64 + 3 * u + m];
      vf[k].y = ca * xrow[64 + 3 * (u + 1) + m];
    }
    float* bv = &ldsI[wave][1 + m][0];
    for (int j = 0; j < 4; ++j) {
      v8f acc = {};
      #pragma unroll
      for (int k = 0; k < 16; ++k) acc = wmma4(vf[k], bfrag_l(fW0v, k, j, lane), acc);
      #pragma unroll
      for (int r = 0; r < 8; ++r) bv[(r + 8 * hi) * LDSTR + j * 16 + n] = acc[r];
    }
  }

  __syncthreads();   // intermediates visible (cross-lane within wave; barrier is safe)

  const float fA = c2 * a * e0;   // midA' and midC' row factor
  const float fD = c2 * a * k3;   // midD' row factor

  // ---------------- out_s = s'@Wscs + midA'@WA + midD'@WD -------------------
  {
    v2f fS[16], fAm[16], fDm[16];
    #pragma unroll
    for (int k = 0; k < 16; ++k) {
      int u = 4 * k + 2 * hi;
      fS[k].x = ca * xrow[u];
      fS[k].y = ca * xrow[u + 1];
      float s0x = bs0[n * LDSTR + u], s0y = bs0[n * LDSTR + u + 1];
      fAm[k].x = fA * trow[u]     * s0x;
      fAm[k].y = fA * trow[u + 1] * s0y;
      float vxa = bv0[n * LDSTR + u], vxb = bv0[n * LDSTR + u + 1];
      float vya = bv1[n * LDSTR + u], vyb = bv1[n * LDSTR + u + 1];
      float vza = bv2[n * LDSTR + u], vzb = bv2[n * LDSTR + u + 1];
      float dax = e1x * vxa + e1y * vya + e1z * vza;
      float day = e1x * vxb + e1y * vyb + e1z * vzb;
      fDm[k].x = fD * trow[192 + u]     * dax;
      fDm[k].y = fD * trow[192 + u + 1] * day;
    }
    for (int j = 0; j < 4; ++j) {
      v8f acc = {};
      #pragma unroll
      for (int k = 0; k < 16; ++k) acc = wmma4(fS[k],  bfrag_g(Wscs, k, j, lane), acc);
      #pragma unroll
      for (int k = 0; k < 16; ++k) acc = wmma4(fAm[k], bfrag_g(WA,   k, j, lane), acc);
      #pragma unroll
      for (int k = 0; k < 16; ++k) acc = wmma4(fDm[k], bfrag_g(WD,   k, j, lane), acc);
      #pragma unroll
      for (int r = 0; r < 8; ++r) {
        int mrow = r0 + r + 8 * hi;
        if (mrow < nrows) out[(size_t)mrow * 256 + j * 16 + n] = acc[r];
      }
    }
  }

  // ---------------- out_v_m = v'_m@Wscv + midB'_m@WB + midC'_m@WC -----------
  for (int m = 0; m < 3; ++m) {
    const float e1m = (m == 0) ? e1x : (m == 1) ? e1y : e1z;
    const float fB  = c2 * a * e1m;
    const float* bvm = &ldsI[wave][1 + m][0];
    v2f fV[16], fBm[16], fCm[16];
    #pragma unroll
    for (int k = 0; k < 16; ++k) {
      int u = 4 * k + 2 * hi;
      fV[k].x = ca * xrow[64 + 3 * u + m];
      fV[k].y = ca * xrow[64 + 3 * (u + 1) + m];
      float s0x = bs0[n * LDSTR + u], s0y = bs0[n * LDSTR + u + 1];
      fBm[k].x = fB * trow[64 + u]     * s0x;
      fBm[k].y = fB * trow[64 + u + 1] * s0y;
      float vmx = bvm[n * LDSTR + u], vmy = bvm[n * LDSTR + u + 1];
      fCm[k].x = fA * trow[128 + u]     * vmx;
      fCm[k].y = fA * trow[128 + u + 1] * vmy;
    }
    for (int j = 0; j < 4; ++j) {
      v8f acc = {};
      #pragma unroll
      for (int k = 0; k < 16; ++k) acc = wmma4(fV[k],  bfrag_l(fWscv, k, j, lane), acc);
      #pragma unroll
      for (int k = 0; k < 16; ++k) acc = wmma4(fBm[k], bfrag_l(fWB,   k, j, lane), acc);
      #pragma unroll
      for (int k = 0; k < 16; ++k) acc = wmma4(fCm[k], bfrag_l(fWC,   k, j, lane), acc);
      #pragma unroll
      for (int r = 0; r < 8; ++r) {
        int mrow = r0 + r + 8 * hi;
        if (mrow < nrows) out[(size_t)mrow * 256 + 64 + (j * 16 + n) * 3 + m] = acc[r];
      }
    }
  }
}

extern "C" void kernel_launch(void* const* d_in, const int* in_sizes, int n_in,
                              void* d_out, int out_size, void* d_ws, size_t ws_size,
                              hipStream_t stream) {
  const float* x1   = (const float*)d_in[0];
  const float* edge = (const float*)d_in[1];
  const float* tpw  = (const float*)d_in[2];
  const float* na   = (const float*)d_in[3];
  const float* W0s  = (const float*)d_in[4];
  const float* W0v  = (const float*)d_in[5];
  const float* Wscs = (const float*)d_in[6];
  const float* Wscv = (const float*)d_in[7];
  const float* WA   = (const float*)d_in[8];
  const float* WD   = (const float*)d_in[9];
  const float* WB   = (const float*)d_in[10];
  const float* WC   = (const float*)d_in[11];

  int nrows  = in_sizes[0] / 256;
  int ntiles = (nrows + TILE_M - 1) / TILE_M;
  int nblocks = (ntiles + WPB - 1) / WPB;

  fused_equivariant_tp<<<nblocks, WPB * 32, 0, stream>>>(
      x1, edge, tpw, na, W0s, W0v, Wscs, Wscv, WA, WD, WB, WC,
      (float*)d_out, nrows, ntiles);
}